// GGNNLayer_71253507441405
// MI455X (gfx1250) — compile-verified
//
#include <hip/hip_runtime.h>
#include <math.h>

#define HIDDEN   128
#define NTYPES   4
#define NNODES   50000
#define NEDGES   400000
#define GRU_N    384          // 3*HIDDEN
#define LDSPITCH 132          // 128 + 4 pad: conflict-free WMMA A reads (16B-aligned pitch)
#define EPB      64           // edges per block (4 M-tiles)
#define NPB      32           // nodes per block in GRU (2 M-tiles)

typedef __attribute__((ext_vector_type(2))) float v2f;
typedef __attribute__((ext_vector_type(8))) float v8f;

// D = A(16x4) * B(4x16) + C, fp32 WMMA (CDNA5 V_WMMA_F32_16X16X4_F32)
__device__ __forceinline__ v8f wmma4_f32(float a0, float a1, float b0, float b1, v8f c) {
    v2f a; a[0] = a0; a[1] = a1;
    v2f b; b[0] = b0; b[1] = b1;
    return __builtin_amdgcn_wmma_f32_16x16x4_f32(
        false, a, false, b, (short)0, c, false, false);
}

// CDNA5 async global->LDS copy (ASYNCcnt-tracked), 16B per lane.
// Generic LDS address low 32 bits == LDS byte offset (ISA aperture rule).
__device__ __forceinline__ void async_ld_b128(void* lds_ptr, const void* gptr) {
    unsigned lds = (unsigned)(unsigned long long)lds_ptr;
    asm volatile("global_load_async_to_lds_b128 %0, %1, off"
                 :: "v"(lds), "v"(gptr) : "memory");
}
__device__ __forceinline__ void wait_async0() {
    asm volatile("s_wait_asynccnt 0x0" ::: "memory");
}
// Guaranteed hardware fp32 atomic (no CAS loop)
__device__ __forceinline__ void atomic_add_f32(float* p, float v) {
    asm volatile("global_atomic_add_f32 %0, %1, off" :: "v"(p), "v"(v) : "memory");
}

__global__ void k_zero(float* __restrict__ p, int n) {
    int i = blockIdx.x * blockDim.x + threadIdx.x;
    if (i < n) p[i] = 0.0f;
}

// -------- edge message kernel: agg[dst] += h[src] @ W[etype] + b[etype] ------
__global__ __launch_bounds__(256) void k_edge_msgs(
    const float* __restrict__ states,   // [NNODES, 128]
    const int*   __restrict__ edges,    // [NEDGES, 3] = (etype, src, dst)
    const float* __restrict__ typeW,    // [4, 128, 128] row-major
    const float* __restrict__ typeB,    // [4, 128]
    float*       __restrict__ agg)      // [NNODES, 128]
{
    __shared__ float Xs[EPB][LDSPITCH];
    __shared__ int   et[EPB], sn[EPB], dn[EPB];

    const int tid = threadIdx.x;
    const int e0  = blockIdx.x * EPB;

    if (tid < EPB) {
        et[tid] = edges[(e0 + tid) * 3 + 0];
        sn[tid] = edges[(e0 + tid) * 3 + 1];
        dn[tid] = edges[(e0 + tid) * 3 + 2];
    }
    __syncthreads();

    // async gather: 64 x 128 f32 = 2048 x b128, 8 per thread
    #pragma unroll
    for (int i = 0; i < 8; ++i) {
        int idx = (tid + i * 256) * 4;           // float index
        int r = idx >> 7, c = idx & 127;
        async_ld_b128(&Xs[r][c], states + (long)sn[r] * HIDDEN + c);
    }
    wait_async0();
    __syncthreads();

    const int lane  = tid & 31;
    const int wave  = tid >> 5;            // 0..7 -> 16-col slice
    const int n16   = lane & 15;           // A row-in-tile / B col-in-tile
    const int kh    = (lane >> 4) << 1;    // K-pair base: 0 or 2
    const int cbase = wave * 16;
    const int colg  = cbase + n16;

    // per-type, per-M-tile row masks
    float sc[NTYPES][4];
    #pragma unroll
    for (int t = 0; t < NTYPES; ++t)
        #pragma unroll
        for (int m = 0; m < 4; ++m)
            sc[t][m] = (et[m * 16 + n16] == t) ? 1.0f : 0.0f;

    v8f acc[4];
    #pragma unroll
    for (int m = 0; m < 4; ++m) acc[m] = (v8f){0.f,0.f,0.f,0.f,0.f,0.f,0.f,0.f};

    #pragma unroll 4
    for (int k = 0; k < HIDDEN; k += 4) {
        const int kk = k + kh;
        float a0[4], a1[4];
        #pragma unroll
        for (int m = 0; m < 4; ++m) {          // A loaded once, reused across 4 types
            a0[m] = Xs[m * 16 + n16][kk];
            a1[m] = Xs[m * 16 + n16][kk + 1];
        }
        #pragma unroll
        for (int t = 0; t < NTYPES; ++t) {
            const float* __restrict__ Wt = typeW + t * HIDDEN * HIDDEN;
            const float b0 = Wt[(kk    ) * HIDDEN + colg];
            const float b1 = Wt[(kk + 1) * HIDDEN + colg];
            #pragma unroll
            for (int m = 0; m < 4; ++m)        // B loaded once, feeds 4 WMMAs
                acc[m] = wmma4_f32(sc[t][m] * a0[m], sc[t][m] * a1[m], b0, b1, acc[m]);
        }
    }

    // C/D layout: VGPR j -> row j (lanes 0-15) or j+8 (lanes 16-31), col = lane%16
    #pragma unroll
    for (int m = 0; m < 4; ++m) {
        #pragma unroll
        for (int j = 0; j < 8; ++j) {
            const int row = m * 16 + j + ((lane >> 4) << 3);
            const float v = acc[m][j] + typeB[et[row] * HIDDEN + colg];
            atomic_add_f32(&agg[(long)dn[row] * HIDDEN + colg], v);
        }
    }
}

// -------- GRU kernel: h' = GRU(x=agg, h) --------------------------------------
__global__ __launch_bounds__(256) void k_gru(
    const float* __restrict__ agg,   // [NNODES, 128]  (x input)
    const float* __restrict__ h,     // [NNODES, 128]
    const float* __restrict__ gk,    // [128, 384]
    const float* __restrict__ grk,   // [128, 384]
    const float* __restrict__ gb,    // [2, 384]
    float*       __restrict__ out)   // [NNODES, 128]
{
    __shared__ float As[NPB][LDSPITCH];
    __shared__ float Hs[NPB][LDSPITCH];

    const int tid = threadIdx.x;
    const int n0  = blockIdx.x * NPB;

    // async stage: 2 x 32 x 128 f32, clamped for the ragged last block
    #pragma unroll
    for (int i = 0; i < 4; ++i) {
        int idx = (tid + i * 256) * 4;
        int r = idx >> 7, c = idx & 127;
        int rg = n0 + r; rg = (rg < NNODES) ? rg : (NNODES - 1);
        async_ld_b128(&As[r][c], agg + (long)rg * HIDDEN + c);
        async_ld_b128(&Hs[r][c], h   + (long)rg * HIDDEN + c);
    }
    wait_async0();
    __syncthreads();

    const int lane  = tid & 31;
    const int wave  = tid >> 5;
    const int n16   = lane & 15;
    const int kh    = (lane >> 4) << 1;
    const int col   = wave * 16 + n16;

    // acc[m][0..5] = zx, rx, cx, zh, rh, ch for M-tile m
    v8f acc[2][6];
    #pragma unroll
    for (int m = 0; m < 2; ++m)
        #pragma unroll
        for (int g = 0; g < 6; ++g) acc[m][g] = (v8f){0.f,0.f,0.f,0.f,0.f,0.f,0.f,0.f};

    #pragma unroll 2
    for (int k = 0; k < HIDDEN; k += 4) {
        const int kk = k + kh;
        float ax0[2], ax1[2], ah0[2], ah1[2];
        #pragma unroll
        for (int m = 0; m < 2; ++m) {
            ax0[m] = As[m * 16 + n16][kk]; ax1[m] = As[m * 16 + n16][kk + 1];
            ah0[m] = Hs[m * 16 + n16][kk]; ah1[m] = Hs[m * 16 + n16][kk + 1];
        }
        const float* __restrict__ K0 = gk  + (long)kk * GRU_N;
        const float* __restrict__ K1 = K0 + GRU_N;
        const float* __restrict__ R0 = grk + (long)kk * GRU_N;
        const float* __restrict__ R1 = R0 + GRU_N;
        const float bz0 = K0[col      ], bz1 = K1[col      ];
        const float br0 = K0[col + 128], br1 = K1[col + 128];
        const float bc0 = K0[col + 256], bc1 = K1[col + 256];
        const float cz0 = R0[col      ], cz1 = R1[col      ];
        const float cr0 = R0[col + 128], cr1 = R1[col + 128];
        const float cc0 = R0[col + 256], cc1 = R1[col + 256];
        #pragma unroll
        for (int m = 0; m < 2; ++m) {          // each B pair feeds 2 WMMAs
            acc[m][0] = wmma4_f32(ax0[m], ax1[m], bz0, bz1, acc[m][0]);
            acc[m][1] = wmma4_f32(ax0[m], ax1[m], br0, br1, acc[m][1]);
            acc[m][2] = wmma4_f32(ax0[m], ax1[m], bc0, bc1, acc[m][2]);
            acc[m][3] = wmma4_f32(ah0[m], ah1[m], cz0, cz1, acc[m][3]);
            acc[m][4] = wmma4_f32(ah0[m], ah1[m], cr0, cr1, acc[m][4]);
            acc[m][5] = wmma4_f32(ah0[m], ah1[m], cc0, cc1, acc[m][5]);
        }
    }

    #pragma unroll
    for (int m = 0; m < 2; ++m) {
        #pragma unroll
        for (int j = 0; j < 8; ++j) {
            const int row = m * 16 + j + ((lane >> 4) << 3);
            const int rg  = n0 + row;
            const float xz = acc[m][0][j] + gb[        col      ];
            const float xr = acc[m][1][j] + gb[        col + 128];
            const float xc = acc[m][2][j] + gb[        col + 256];
            const float hz = acc[m][3][j] + gb[GRU_N + col      ];
            const float hr = acc[m][4][j] + gb[GRU_N + col + 128];
            const float hc = acc[m][5][j] + gb[GRU_N + col + 256];
            const float z  = 1.0f / (1.0f + expf(-(xz + hz)));
            const float r  = 1.0f / (1.0f + expf(-(xr + hr)));
            const float ct = tanhf(xc + r * hc);
            const float hv = Hs[row][col];
            if (rg < NNODES)
                out[(long)rg * HIDDEN + col] = z * hv + (1.0f - z) * ct;
        }
    }
}

extern "C" void kernel_launch(void* const* d_in, const int* in_sizes, int n_in,
                              void* d_out, int out_size, void* d_ws, size_t ws_size,
                              hipStream_t stream) {
    const float* states = (const float*)d_in[0];   // [50000,128]
    const int*   edges  = (const int*)  d_in[1];   // [400000,3]
    const float* typeW  = (const float*)d_in[2];   // [1,4,128,128]
    const float* typeB  = (const float*)d_in[3];   // [1,4,128]
    const float* gk     = (const float*)d_in[4];   // [1,128,384]
    const float* grk    = (const float*)d_in[5];   // [1,128,384]
    const float* gb     = (const float*)d_in[6];   // [1,2,384]
    float* out    = (float*)d_out;
    float* aggbuf = (float*)d_ws;                  // 50000*128 f32 scratch

    const int nagg = NNODES * HIDDEN;
    k_zero<<<(nagg + 255) / 256, 256, 0, stream>>>(aggbuf, nagg);
    k_edge_msgs<<<NEDGES / EPB, 256, 0, stream>>>(states, edges, typeW, typeB, aggbuf);
    k_gru<<<(NNODES + NPB - 1) / NPB, 256, 0, stream>>>(aggbuf, states, gk, grk, gb, out);
}